// SetAbstraction_75419625717816
// MI455X (gfx1250) — compile-verified
//
#include <hip/hip_runtime.h>
#include <hip/hip_bf16.h>

// Problem constants (from reference)
#define BATCH   16
#define NPTS    8192
#define NPOINT  1024
#define NSAMP   32
#define INCH    64
#define CK      67      // IN_CH + 3
#define KPAD    68      // padded K (multiple of 4)
#define OC      128     // 2*IN_CH

typedef __attribute__((ext_vector_type(2))) float v2f;
typedef __attribute__((ext_vector_type(8))) float v8f;

// ---------------------------------------------------------------------------
// Kernel 1: farthest point sampling. One block (1024 threads) per batch.
// Each thread owns 8 points in registers. Matches jnp.argmax tie-breaking
// (smallest index among equal maxima).
// ---------------------------------------------------------------------------
__global__ __launch_bounds__(1024) void fps_kernel(const float* __restrict__ coor,
                                                   int* __restrict__ fps_idx) {
  const int b = blockIdx.x;
  const int t = threadIdx.x;
  const float* cb = coor + (size_t)b * 3 * NPTS;

  float x[8], y[8], z[8], dist[8];
  for (int i = 0; i < 8; ++i) {
    int n = i * 1024 + t;
    x[i] = cb[n];
    y[i] = cb[NPTS + n];
    z[i] = cb[2 * NPTS + n];
    dist[i] = 1e10f;
  }

  __shared__ float s_c[3];
  __shared__ float s_wv[32];
  __shared__ int   s_wi[32];
  __shared__ int   s_fi;

  const int wid = t >> 5, lane = t & 31;
  int fi = 0;

  for (int j = 0; j < NPOINT; ++j) {
    if (t == 0) fps_idx[b * NPOINT + j] = fi;
    // owner of current farthest point publishes its coords
    if ((fi & 1023) == t) {
      int slot = fi >> 10;
      s_c[0] = x[slot]; s_c[1] = y[slot]; s_c[2] = z[slot];
    }
    __syncthreads();
    float cx = s_c[0], cy = s_c[1], cz = s_c[2];

    float bv = -1.0f;
    int   bi = 0x7fffffff;
    for (int i = 0; i < 8; ++i) {
      float dx = x[i] - cx, dy = y[i] - cy, dz = z[i] - cz;
      float d  = dx * dx + dy * dy + dz * dz;
      float nd = fminf(dist[i], d);
      dist[i] = nd;
      int n = i * 1024 + t;
      if (nd > bv || (nd == bv && n < bi)) { bv = nd; bi = n; }
    }
    // wave32 argmax reduce
    for (int m = 16; m >= 1; m >>= 1) {
      float ov = __shfl_xor(bv, m, 32);
      int   oi = __shfl_xor(bi, m, 32);
      if (ov > bv || (ov == bv && oi < bi)) { bv = ov; bi = oi; }
    }
    if (lane == 0) { s_wv[wid] = bv; s_wi[wid] = bi; }
    __syncthreads();
    if (wid == 0) {
      bv = s_wv[lane]; bi = s_wi[lane];
      for (int m = 16; m >= 1; m >>= 1) {
        float ov = __shfl_xor(bv, m, 32);
        int   oi = __shfl_xor(bi, m, 32);
        if (ov > bv || (ov == bv && oi < bi)) { bv = ov; bi = oi; }
      }
      if (lane == 0) s_fi = bi;
    }
    __syncthreads();
    fi = s_fi;
  }
}

// ---------------------------------------------------------------------------
// Kernel 2: hybrid KNN / ball query. One block (256 threads) per centroid.
// Exact top-32 by repeated argmin over an LDS distance array, with the
// reference radius substitution (d > R^2 -> closest index). Also emits the
// new_coor output (B,3,S).
// ---------------------------------------------------------------------------
__global__ __launch_bounds__(256) void knn_kernel(const float* __restrict__ coor,
                                                  const int* __restrict__ fps_idx,
                                                  int* __restrict__ gidx,
                                                  float* __restrict__ out_coor) {
  const int blk = blockIdx.x;          // == centroid id
  const int b = blk >> 10, s = blk & 1023;
  const int t = threadIdx.x;
  const float* cb = coor + (size_t)b * 3 * NPTS;

  const int ci = fps_idx[b * NPOINT + s];
  const float cx = cb[ci], cy = cb[NPTS + ci], cz = cb[2 * NPTS + ci];
  if (t < 3)
    out_coor[((size_t)b * 3 + t) * NPOINT + s] = (t == 0) ? cx : ((t == 1) ? cy : cz);

  __shared__ float sd[NPTS];
  __shared__ float swv[8];
  __shared__ int   swi[8];
  __shared__ float sel_d[NSAMP];
  __shared__ int   sel_i[NSAMP];

  for (int i = 0; i < 32; ++i) {
    int n = i * 256 + t;
    float dx = cb[n] - cx, dy = cb[NPTS + n] - cy, dz = cb[2 * NPTS + n] - cz;
    sd[n] = dx * dx + dy * dy + dz * dz;
  }
  __syncthreads();

  const int wid = t >> 5, lane = t & 31;
  for (int r = 0; r < NSAMP; ++r) {
    float bv = 3.402823466e38f;
    int   bi = 0x7fffffff;
    for (int i = 0; i < 32; ++i) {
      int n = i * 256 + t;
      float v = sd[n];
      if (v < bv || (v == bv && n < bi)) { bv = v; bi = n; }
    }
    for (int m = 16; m >= 1; m >>= 1) {
      float ov = __shfl_xor(bv, m, 32);
      int   oi = __shfl_xor(bi, m, 32);
      if (ov < bv || (ov == bv && oi < bi)) { bv = ov; bi = oi; }
    }
    if (lane == 0) { swv[wid] = bv; swi[wid] = bi; }
    __syncthreads();
    if (t == 0) {
      float mv = swv[0]; int mi = swi[0];
      for (int w = 1; w < 8; ++w) {
        float v = swv[w]; int ii = swi[w];
        if (v < mv || (v == mv && ii < mi)) { mv = v; mi = ii; }
      }
      sel_d[r] = mv; sel_i[r] = mi;
      sd[mi] = 3.402823466e38f;   // remove from candidates
    }
    __syncthreads();
  }

  if (t < NSAMP) {
    float d = sel_d[t];
    int   i = sel_i[t];
    gidx[(size_t)blk * NSAMP + t] = (d > 1.0f) ? sel_i[0] : i;  // RADIUS^2 = 1
  }
}

// ---------------------------------------------------------------------------
// Kernel 3: gather + 1x1 conv (WMMA f32 16x16x4) + ReLU + group max-pool.
// 256 threads = 8 waves; each wave owns one centroid: (32x68)x(68x128) GEMM
// via 2 M-tiles x 8 N-tiles x 17 K-steps = 272 v_wmma_f32_16x16x4_f32.
// W is staged (transposed, K-padded) in LDS once per block.
// ---------------------------------------------------------------------------
__global__ __launch_bounds__(256) void conv_kernel(const float* __restrict__ coor,
                                                   const float* __restrict__ fea,
                                                   const float* __restrict__ W,
                                                   const int* __restrict__ fps_idx,
                                                   const int* __restrict__ gidx,
                                                   float* __restrict__ out_fea) {
  __shared__ float Wlds[KPAD * OC];   // Wlds[k*128 + o] = W[o][k] (zero-padded k=67)
  const int t = threadIdx.x;
  for (int i = t; i < KPAD * OC; i += 256) {
    int k = i >> 7, o = i & 127;
    Wlds[i] = (k < CK) ? W[o * CK + k] : 0.0f;
  }
  __syncthreads();

  const int wid = t >> 5, lane = t & 31;
  const int cid = blockIdx.x * 8 + wid;          // centroid id, 0..16383
  const int b = cid >> 10, s = cid & 1023;
  const float* cb = coor + (size_t)b * 3 * NPTS;
  const float* fb = fea + (size_t)b * INCH * NPTS;

  const int ci = fps_idx[cid];
  const float c0 = cb[ci], c1 = cb[NPTS + ci], c2 = cb[2 * NPTS + ci];

  const int ml = lane & 15;                       // row within 16-row tile / col
  const int gi_lo = gidx[(size_t)cid * NSAMP + ml];        // rows 0..15
  const int gi_hi = gidx[(size_t)cid * NSAMP + 16 + ml];   // rows 16..31
  const int koff = (lane < 16) ? 0 : 2;           // K split across lane halves

  v8f acc[2][8];
  for (int mt = 0; mt < 2; ++mt)
    for (int nt = 0; nt < 8; ++nt)
      for (int i = 0; i < 8; ++i) acc[mt][nt][i] = 0.0f;

  for (int ks = 0; ks < KPAD / 4; ++ks) {
    const int k0 = ks * 4 + koff;                 // even: 0,2,...,66
    const int k1 = k0 + 1;
    v2f a[2];
    for (int mt = 0; mt < 2; ++mt) {
      const int gi = mt ? gi_hi : gi_lo;
      float a0, a1;
      if (k0 < INCH)        a0 = fb[(size_t)k0 * NPTS + gi];
      else if (k0 == 64)    a0 = cb[gi] - c0;
      else /* k0 == 66 */   a0 = cb[2 * NPTS + gi] - c2;
      if (k1 < INCH)        a1 = fb[(size_t)k1 * NPTS + gi];
      else if (k1 == 65)    a1 = cb[NPTS + gi] - c1;
      else /* k1 == 67 */   a1 = 0.0f;            // K padding
      a[mt].x = a0; a[mt].y = a1;
    }
    for (int nt = 0; nt < 8; ++nt) {
      v2f bb;
      bb.x = Wlds[k0 * OC + nt * 16 + ml];
      bb.y = Wlds[k1 * OC + nt * 16 + ml];
      acc[0][nt] = __builtin_amdgcn_wmma_f32_16x16x4_f32(
          false, a[0], false, bb, (short)0, acc[0][nt], false, false);
      acc[1][nt] = __builtin_amdgcn_wmma_f32_16x16x4_f32(
          false, a[1], false, bb, (short)0, acc[1][nt], false, false);
    }
  }

  // ReLU + max over 32 group rows. Lane holds rows {8*(lane/16)+v} of each
  // 16-row tile, col = lane%16; shfl_xor(16) merges the complementary rows.
  for (int nt = 0; nt < 8; ++nt) {
    float v = 0.0f;                                // ReLU floor
    for (int mt = 0; mt < 2; ++mt)
      for (int i = 0; i < 8; ++i) v = fmaxf(v, acc[mt][nt][i]);
    v = fmaxf(v, __shfl_xor(v, 16, 32));
    if (lane < 16) {
      int o = nt * 16 + lane;
      out_fea[((size_t)b * OC + o) * NPOINT + s] = v;
    }
  }
}

// ---------------------------------------------------------------------------
extern "C" void kernel_launch(void* const* d_in, const int* in_sizes, int n_in,
                              void* d_out, int out_size, void* d_ws, size_t ws_size,
                              hipStream_t stream) {
  const float* coor = (const float*)d_in[0];   // (16,3,8192)
  const float* fea  = (const float*)d_in[1];   // (16,64,8192)
  const float* W    = (const float*)d_in[2];   // (128,67)

  float* out      = (float*)d_out;
  float* out_coor = out;                        // (16,3,1024)
  float* out_fea  = out + (size_t)BATCH * 3 * NPOINT;  // (16,128,1024)

  int* fps_idx = (int*)d_ws;                    // 16*1024 ints
  int* gidx    = fps_idx + BATCH * NPOINT;      // 16*1024*32 ints

  fps_kernel<<<BATCH, 1024, 0, stream>>>(coor, fps_idx);
  knn_kernel<<<BATCH * NPOINT, 256, 0, stream>>>(coor, fps_idx, gidx, out_coor);
  conv_kernel<<<BATCH * NPOINT / 8, 256, 0, stream>>>(coor, fea, W, fps_idx, gidx, out_fea);

  (void)in_sizes; (void)n_in; (void)out_size; (void)ws_size;
}